// Evaluation_86955907875354
// MI455X (gfx1250) — compile-verified
//
#include <hip/hip_runtime.h>
#include <math.h>

// Problem constants (match reference)
#define Bc 2
#define Vc 4
#define Cc 64
#define Dc 16
#define Hc 160
#define Wc 200
#define Gc 8
#define NB 9
#define HWc (Hc*Wc)          // 32000
#define DHWc (Dc*HWc)        // 512000
#define EPSc 1e-5f

typedef __attribute__((ext_vector_type(2))) float v2f;
typedef __attribute__((ext_vector_type(8))) float v8f;

// ---- workspace layout (float offsets) ----
constexpr size_t WS_PROJ   = 0;                                   // V*B*12 = 96
constexpr size_t WS_WF_PW  = 128;                                 // 512
constexpr size_t WS_WF_SN  = 768;                                 // 512
constexpr size_t WS_SIMV   = 2048;                                // B*G*DHW = 8,192,000
constexpr size_t WS_SIMSUM = WS_SIMV   + (size_t)Bc*Gc*DHWc;
constexpr size_t WS_WSUM   = WS_SIMSUM + (size_t)Bc*Gc*DHWc;      // B*HW
constexpr size_t WS_PSC    = WS_WSUM   + (size_t)Bc*HWc;          // B*DHW (reused for sn score)
constexpr size_t WS_SRAW   = WS_PSC    + (size_t)Bc*DHWc;         // B*DHW
// total ~18.5M floats ~= 74 MB

// folded-weight block layout (within 512-float block)
#define WF_W0 0      // 16x8
#define WF_C0 128    // 16
#define WF_W1 144    // 16x16 (rows 8..15 zero)
#define WF_C1 400    // 16
#define WF_W2 416    // 8
#define WF_B2 424    // 1

// output layout: depth (B*HW) | score (B*D*HW) | view_weights (B*V*HW)
constexpr size_t OUT_SCORE = (size_t)Bc*HWc;              // 64000
constexpr size_t OUT_VW    = OUT_SCORE + (size_t)Bc*DHWc; // 1,088,000

__device__ __forceinline__ int imin(int a, int b){ return a<b?a:b; }
__device__ __forceinline__ int imax(int a, int b){ return a>b?a:b; }

// ---------------- setup: 4x4 inverse, projection compose, BN folding ----------------
__device__ void inv4(const float* m, float* inv){
  float a[4][8];
  for(int i=0;i<4;i++){ for(int j=0;j<4;j++){ a[i][j]=m[i*4+j]; a[i][j+4]=(i==j)?1.f:0.f; } }
  for(int col=0; col<4; col++){
    int piv=col; float best=fabsf(a[col][col]);
    for(int rw=col+1; rw<4; rw++){ float v=fabsf(a[rw][col]); if(v>best){best=v;piv=rw;} }
    if(piv!=col){ for(int j=0;j<8;j++){ float t=a[col][j]; a[col][j]=a[piv][j]; a[piv][j]=t; } }
    float d=1.f/a[col][col];
    for(int j=0;j<8;j++) a[col][j]*=d;
    for(int rw=0;rw<4;rw++){ if(rw==col) continue; float f=a[rw][col];
      for(int j=0;j<8;j++) a[rw][j]-=f*a[col][j]; }
  }
  for(int i=0;i<4;i++) for(int j=0;j<4;j++) inv[i*4+j]=a[i][j+4];
}

__device__ void fold_net(const float* w0,const float* bn0,const float* w1,const float* bn1,
                         const float* w2,const float* b2, float* wf){
  // bn rows: [gamma, beta, mean, var] each of length cch
  for(int j=0;j<16;j++){
    float g=bn0[j], bb=bn0[16+j], mm=bn0[32+j], vv=bn0[48+j];
    float aa = g * rsqrtf(vv + EPSc);
    for(int k=0;k<8;k++) wf[WF_W0 + j*8+k] = aa * w0[j*8+k];
    wf[WF_C0 + j] = bb - mm*aa;
  }
  for(int j=0;j<16;j++){
    if(j<8){
      float g=bn1[j], bb=bn1[8+j], mm=bn1[16+j], vv=bn1[24+j];
      float aa = g * rsqrtf(vv + EPSc);
      for(int k=0;k<16;k++) wf[WF_W1 + j*16+k] = aa * w1[j*16+k];
      wf[WF_C1 + j] = bb - mm*aa;
    } else {
      for(int k=0;k<16;k++) wf[WF_W1 + j*16+k] = 0.f;
      wf[WF_C1 + j] = 0.f;
    }
  }
  for(int k=0;k<8;k++) wf[WF_W2+k]=w2[k];
  wf[WF_B2]=b2[0];
}

__global__ void k_setup(const float* ref_proj, const float* src_projs,
                        const float* pw_w0,const float* pw_bn0,const float* pw_w1,const float* pw_bn1,
                        const float* pw_w2,const float* pw_b2,
                        const float* sn_w0,const float* sn_bn0,const float* sn_w1,const float* sn_bn1,
                        const float* sn_w2,const float* sn_b2,
                        float* ws){
  int t = threadIdx.x;
  if(t==0) fold_net(pw_w0,pw_bn0,pw_w1,pw_bn1,pw_w2,pw_b2, ws+WS_WF_PW);
  if(t==1) fold_net(sn_w0,sn_bn0,sn_w1,sn_bn1,sn_w2,sn_b2, ws+WS_WF_SN);
  if(t>=2 && t<2+Vc*Bc){
    int idx=t-2; int v=idx/Bc, b=idx%Bc;
    float inv[16], proj[16];
    inv4(ref_proj + b*16, inv);
    const float* sp = src_projs + ((size_t)v*Bc + b)*16;
    for(int i=0;i<4;i++) for(int j=0;j<4;j++){
      float s=0.f; for(int k=0;k<4;k++) s += sp[i*4+k]*inv[k*4+j]; proj[i*4+j]=s; }
    float* o = ws + WS_PROJ + (size_t)idx*12;
    for(int i=0;i<3;i++){ for(int j=0;j<3;j++) o[i*3+j]=proj[i*4+j]; o[9+i]=proj[i*4+3]; }
  }
}

__global__ void k_init(float* ws){
  size_t i = (size_t)blockIdx.x*blockDim.x + threadIdx.x;
  if(i < (size_t)Bc*Gc*DHWc) ws[WS_SIMSUM + i] = 0.f;
  if(i < (size_t)Bc*HWc)     ws[WS_WSUM + i] = 1e-5f;
}

// ---------------- warp + group correlation (gather-bound; L2-resident src) ----------------
__global__ __launch_bounds__(256) void k_warpcorr(
    const float* __restrict__ srcv,   // (B,C,H,W) for this view
    const float* __restrict__ ref,    // (B,C,H,W)
    const float* __restrict__ depth,  // (B,D,H,W)
    const float* __restrict__ projvb, // ws proj for this view: B*12
    float* __restrict__ simv){        // (B,G,D,HW)
  int i = blockIdx.x*blockDim.x + threadIdx.x;
  if(i >= Bc*DHWc) return;
  int b = i / DHWc; int r = i - b*DHWc;
  int hw = r % HWc;
  int y = hw / Wc, x = hw - y*Wc;
  const float* P = projvb + b*12;
  float fx=(float)x, fy=(float)y;
  float rx = P[0]*fx + P[1]*fy + P[2];
  float ry = P[3]*fx + P[4]*fy + P[5];
  float rz = P[6]*fx + P[7]*fy + P[8];
  float dep = depth[(size_t)b*DHWc + r];
  float px3 = rx*dep + P[9];
  float py3 = ry*dep + P[10];
  float pz3 = rz*dep + P[11];
  bool neg = pz3 <= 0.001f;
  float zs = neg?1.f:pz3;
  float xp = neg ? (float)Wc : px3/zs;   // align_corners=True: grid->pixel is identity
  float yp = neg ? (float)Hc : py3/zs;
  float pxc = fminf(fmaxf(xp, -1.f), (float)Wc);
  float pyc = fminf(fmaxf(yp, -1.f), (float)Hc);
  float x0f = floorf(pxc), y0f = floorf(pyc);
  float wx1 = pxc - x0f, wy1 = pyc - y0f;
  float wx0 = 1.f-wx1, wy0 = 1.f-wy1;
  int x0=(int)x0f, y0=(int)y0f, x1=x0+1, y1=y0+1;
  float vx0 = (x0>=0 && x0<Wc)?1.f:0.f;
  float vx1 = (x1>=0 && x1<Wc)?1.f:0.f;
  float vy0 = (y0>=0 && y0<Hc)?1.f:0.f;
  float vy1 = (y1>=0 && y1<Hc)?1.f:0.f;
  int x0c=imin(imax(x0,0),Wc-1), x1c=imin(imax(x1,0),Wc-1);
  int y0c=imin(imax(y0,0),Hc-1), y1c=imin(imax(y1,0),Hc-1);
  float w00=wy0*wx0*vy0*vx0, w01=wy0*wx1*vy0*vx1;
  float w10=wy1*wx0*vy1*vx0, w11=wy1*wx1*vy1*vx1;
  int o00=y0c*Wc+x0c, o01=y0c*Wc+x1c, o10=y1c*Wc+x0c, o11=y1c*Wc+x1c;
  const float* sb = srcv + (size_t)b*Cc*HWc;
  const float* rb = ref  + (size_t)b*Cc*HWc + hw;
  float sim[Gc];
  #pragma unroll
  for(int g=0;g<Gc;g++) sim[g]=0.f;
  #pragma unroll 8
  for(int c=0;c<Cc;c++){
    const float* p = sb + (size_t)c*HWc;
    float wv = w00*p[o00] + w01*p[o01] + w10*p[o10] + w11*p[o11];
    sim[c>>3] += wv * rb[(size_t)c*HWc];
  }
  float* so = simv + (size_t)b*Gc*DHWc + r;
  #pragma unroll
  for(int g=0;g<Gc;g++) so[(size_t)g*DHWc] = sim[g]*0.125f;
}

// ---------------- WMMA per-voxel MLP: 8 -> 16 -> 8 -> 1, 16 voxels per wave-tile ----------------
__global__ __launch_bounds__(256) void k_mlp(
    const float* __restrict__ x,     // (B,G,D,HW)
    const float* __restrict__ wf,    // folded weight block (512 floats)
    const float* __restrict__ wsum,  // null => no input division
    float* __restrict__ out,         // (B,D,HW)
    int sigmoid_flag){
  int lane = threadIdx.x & 31;
  int gw = blockIdx.x*(blockDim.x>>5) + (threadIdx.x>>5);
  int nw = gridDim.x*(blockDim.x>>5);
  int jn = lane & 15, hi = lane >> 4;

  // A tiles: 16x4 f32 layout -> lane holds row M=jn, VGPR0/1 = K in {2*hi, 2*hi+1}
  v2f A0, A1, A2[4];
  A0.x = wf[WF_W0 + jn*8 + 2*hi];       A0.y = wf[WF_W0 + jn*8 + 2*hi + 1];
  A1.x = wf[WF_W0 + jn*8 + 4 + 2*hi];   A1.y = wf[WF_W0 + jn*8 + 5 + 2*hi];
  #pragma unroll
  for(int kk=0;kk<4;kk++){
    A2[kk].x = wf[WF_W1 + jn*16 + 4*kk + 2*hi];
    A2[kk].y = wf[WF_W1 + jn*16 + 4*kk + 2*hi + 1];
  }
  float c0b[8], c1b[8], w2v[8];
  #pragma unroll
  for(int r2=0;r2<8;r2++){
    c0b[r2] = wf[WF_C0 + r2 + 8*hi];   // C/D: vgpr r holds row r + 8*hi
    c1b[r2] = wf[WF_C1 + r2 + 8*hi];
    w2v[r2] = wf[WF_W2 + r2];
  }
  float b2v = wf[WF_B2];
  bool hib = (hi != 0);

  const int ntiles = (Bc*DHWc)/16;
  for(int t = gw; t < ntiles; t += nw){
    int n = t*16 + jn;                // this lane's voxel column
    int b = n / DHWc; int r = n - b*DHWc;
    const float* xb = x + (size_t)b*Gc*DHWc + r;
    // B tiles: lane holds column N=jn, K in {2*hi, 2*hi+1}
    v2f B0, B1;
    B0.x = xb[(size_t)(2*hi)  *DHWc];  B0.y = xb[(size_t)(2*hi+1)*DHWc];
    B1.x = xb[(size_t)(4+2*hi)*DHWc];  B1.y = xb[(size_t)(5+2*hi)*DHWc];
    if(wsum){ // uniform branch: only the similarity-net pass divides by wsum
      int hw = r % HWc;
      float invw = 1.f / wsum[b*HWc + hw];
      B0.x *= invw; B0.y *= invw; B1.x *= invw; B1.y *= invw;
    }

    v8f c;
    #pragma unroll
    for(int r2=0;r2<8;r2++) c[r2] = c0b[r2];
    c = __builtin_amdgcn_wmma_f32_16x16x4_f32(false, A0, false, B0, (short)0, c, false, false);
    c = __builtin_amdgcn_wmma_f32_16x16x4_f32(false, A1, false, B1, (short)0, c, false, false);
    float y1[8];
    #pragma unroll
    for(int r2=0;r2<8;r2++) y1[r2] = fmaxf(c[r2], 0.f);   // relu(layer1)

    // D-layout -> B-layout for layer2: swap half-waves for cross-half K rows,
    // then per-lane select (v_cndmask, no EXEC divergence).
    float s0=__shfl_xor(y1[0],16,32), s1=__shfl_xor(y1[1],16,32);
    float s2=__shfl_xor(y1[2],16,32), s3=__shfl_xor(y1[3],16,32);
    float s4=__shfl_xor(y1[4],16,32), s5=__shfl_xor(y1[5],16,32);
    float s6=__shfl_xor(y1[6],16,32), s7=__shfl_xor(y1[7],16,32);
    v2f Bk0,Bk1,Bk2,Bk3;
    Bk0.x = hib ? s2    : y1[0];  Bk0.y = hib ? s3    : y1[1];
    Bk1.x = hib ? s6    : y1[4];  Bk1.y = hib ? s7    : y1[5];
    Bk2.x = hib ? y1[2] : s0;     Bk2.y = hib ? y1[3] : s1;
    Bk3.x = hib ? y1[6] : s4;     Bk3.y = hib ? y1[7] : s5;

    v8f c2;
    #pragma unroll
    for(int r2=0;r2<8;r2++) c2[r2] = c1b[r2];
    c2 = __builtin_amdgcn_wmma_f32_16x16x4_f32(false, A2[0], false, Bk0, (short)0, c2, false, false);
    c2 = __builtin_amdgcn_wmma_f32_16x16x4_f32(false, A2[1], false, Bk1, (short)0, c2, false, false);
    c2 = __builtin_amdgcn_wmma_f32_16x16x4_f32(false, A2[2], false, Bk2, (short)0, c2, false, false);
    c2 = __builtin_amdgcn_wmma_f32_16x16x4_f32(false, A2[3], false, Bk3, (short)0, c2, false, false);

    if(hi==0){ // lanes 0..15 hold rows 0..7 (all 8 layer2 outputs) of their column
      float s = b2v;
      #pragma unroll
      for(int r2=0;r2<8;r2++) s += w2v[r2]*fmaxf(c2[r2],0.f);
      if(sigmoid_flag) s = 1.f/(1.f+__expf(-s));
      out[n] = s;
    }
  }
}

// ---------------- view weight = max over D; update wsum ----------------
__global__ void k_vwmax(const float* __restrict__ psc, float* __restrict__ wsum,
                        float* __restrict__ vw_out, int v){
  int i = blockIdx.x*blockDim.x + threadIdx.x;
  if(i >= Bc*HWc) return;
  int b = i / HWc; int hw = i - b*HWc;
  const float* p = psc + (size_t)b*DHWc + hw;
  float m = p[0];
  #pragma unroll
  for(int d=1; d<Dc; d++) m = fmaxf(m, p[(size_t)d*HWc]);
  vw_out[((size_t)b*Vc + v)*HWc + hw] = m;
  wsum[i] += m;
}

// ---------------- sim_sum += sim_v * vw ----------------
__global__ void k_accum(const float* __restrict__ simv, const float* __restrict__ vw,
                        int v, float* __restrict__ simsum){
  size_t i = (size_t)blockIdx.x*blockDim.x + threadIdx.x;
  if(i >= (size_t)Bc*Gc*DHWc) return;
  int b = (int)(i / ((size_t)Gc*DHWc));
  int hw = (int)(i % HWc);
  simsum[i] += simv[i] * vw[((size_t)b*Vc + v)*HWc + hw];
}

// ---------------- 9-neighbor border-bilinear resample + weighted sum ----------------
__global__ __launch_bounds__(256) void k_neigh(
    const float* __restrict__ score0, const float* __restrict__ grid,
    const float* __restrict__ wgt, float* __restrict__ sraw){
  int i = blockIdx.x*blockDim.x + threadIdx.x;
  if(i >= Bc*DHWc) return;
  int b = i / DHWc; int r = i - b*DHWc;
  int d = r / HWc; int hw = r - d*HWc;
  int y = hw / Wc, x = hw - y*Wc;
  const float* img = score0 + (size_t)b*DHWc + (size_t)d*HWc;
  float acc = 0.f;
  #pragma unroll
  for(int k=0;k<NB;k++){
    size_t gi = (((size_t)b*(NB*Hc) + (size_t)k*Hc + y)*Wc + x)*2;
    float gx = grid[gi], gy = grid[gi+1];
    float px = ((gx+1.f)*Wc - 1.f)*0.5f;      // align_corners=False
    float py = ((gy+1.f)*Hc - 1.f)*0.5f;
    px = fminf(fmaxf(px, 0.f), (float)(Wc-1)); // border padding
    py = fminf(fmaxf(py, 0.f), (float)(Hc-1));
    float x0f = floorf(px), y0f = floorf(py);
    float wx1 = px-x0f, wy1 = py-y0f;
    int x0=(int)x0f, y0=(int)y0f;
    int x1=imin(x0+1,Wc-1), y1=imin(y0+1,Hc-1);
    float v00=img[y0*Wc+x0], v01=img[y0*Wc+x1];
    float v10=img[y1*Wc+x0], v11=img[y1*Wc+x1];
    float sv = (1.f-wy1)*((1.f-wx1)*v00 + wx1*v01) + wy1*((1.f-wx1)*v10 + wx1*v11);
    acc += sv * wgt[((((size_t)b*Dc + d)*NB + k)*Hc + y)*Wc + x];
  }
  sraw[i] = acc;
}

// ---------------- softmax over D + expected depth ----------------
__global__ void k_softmax(const float* __restrict__ sraw, const float* __restrict__ dsample,
                          float* __restrict__ out){
  int i = blockIdx.x*blockDim.x + threadIdx.x;
  if(i >= Bc*HWc) return;
  int b = i / HWc; int hw = i - b*HWc;
  const float* p = sraw + (size_t)b*DHWc + hw;
  float vals[Dc], m = -1e30f;
  #pragma unroll
  for(int d=0;d<Dc;d++){ vals[d] = p[(size_t)d*HWc]; m = fmaxf(m, vals[d]); }
  float sum=0.f;
  #pragma unroll
  for(int d=0;d<Dc;d++){ vals[d] = __expf(vals[d]-m); sum += vals[d]; }
  float inv = 1.f/sum;
  const float* ds = dsample + (size_t)b*DHWc + hw;
  float* score = out + OUT_SCORE;
  float depth = 0.f;
  #pragma unroll
  for(int d=0;d<Dc;d++){
    float pr = vals[d]*inv;
    score[(size_t)b*DHWc + (size_t)d*HWc + hw] = pr;
    depth += ds[(size_t)d*HWc]*pr;
  }
  out[i] = depth;
}

extern "C" void kernel_launch(void* const* d_in, const int* in_sizes, int n_in,
                              void* d_out, int out_size, void* d_ws, size_t ws_size,
                              hipStream_t stream){
  (void)in_sizes; (void)n_in; (void)out_size; (void)ws_size;
  const float* ref_feature  = (const float*)d_in[0];
  const float* src_features = (const float*)d_in[1];
  const float* ref_proj     = (const float*)d_in[2];
  const float* src_projs    = (const float*)d_in[3];
  const float* depth_sample = (const float*)d_in[4];
  const float* grid         = (const float*)d_in[7];
  const float* weight       = (const float*)d_in[8];
  const float* pw_w0  = (const float*)d_in[9];
  const float* pw_bn0 = (const float*)d_in[10];
  const float* pw_w1  = (const float*)d_in[11];
  const float* pw_bn1 = (const float*)d_in[12];
  const float* pw_w2  = (const float*)d_in[13];
  const float* pw_b2  = (const float*)d_in[14];
  const float* sn_w0  = (const float*)d_in[15];
  const float* sn_bn0 = (const float*)d_in[16];
  const float* sn_w1  = (const float*)d_in[17];
  const float* sn_bn1 = (const float*)d_in[18];
  const float* sn_w2  = (const float*)d_in[19];
  const float* sn_b2  = (const float*)d_in[20];
  float* out = (float*)d_out;
  float* ws  = (float*)d_ws;

  k_setup<<<1,64,0,stream>>>(ref_proj, src_projs,
                             pw_w0,pw_bn0,pw_w1,pw_bn1,pw_w2,pw_b2,
                             sn_w0,sn_bn0,sn_w1,sn_bn1,sn_w2,sn_b2, ws);
  k_init<<<32000,256,0,stream>>>(ws);

  for(int v=0; v<Vc; v++){
    k_warpcorr<<<4000,256,0,stream>>>(src_features + (size_t)v*Bc*Cc*HWc,
                                      ref_feature, depth_sample,
                                      ws + WS_PROJ + (size_t)v*Bc*12,
                                      ws + WS_SIMV);
    k_mlp<<<2000,256,0,stream>>>(ws + WS_SIMV, ws + WS_WF_PW, nullptr, ws + WS_PSC, 1);
    k_vwmax<<<250,256,0,stream>>>(ws + WS_PSC, ws + WS_WSUM, out + OUT_VW, v);
    k_accum<<<32000,256,0,stream>>>(ws + WS_SIMV, out + OUT_VW, v, ws + WS_SIMSUM);
  }

  // similarity net on sim_sum/wsum -> raw score volume (reuse PSC buffer)
  k_mlp<<<2000,256,0,stream>>>(ws + WS_SIMSUM, ws + WS_WF_SN, ws + WS_WSUM, ws + WS_PSC, 0);
  k_neigh<<<4000,256,0,stream>>>(ws + WS_PSC, grid, weight, ws + WS_SRAW);
  k_softmax<<<250,256,0,stream>>>(ws + WS_SRAW, depth_sample, out);
}